// FalconGNN_88347477278819
// MI455X (gfx1250) — compile-verified
//
#include <hip/hip_runtime.h>
#include <math.h>

#define NNODES 100000
#define NEDGES 300000
#define DNODE  64
#define HDIM   256

typedef float v2f __attribute__((ext_vector_type(2)));
typedef float v8f __attribute__((ext_vector_type(8)));

// fp32 WMMA: V_WMMA_F32_16X16X4_F32  (D[16x16] = A[16x4] * B[4x16] + C)
#if defined(__gfx1250__) && __has_builtin(__builtin_amdgcn_wmma_f32_16x16x4_f32)
#define HAVE_WMMA_F32 1
#endif

#ifdef HAVE_WMMA_F32
static __device__ __forceinline__ v8f wmma4(v2f a, v2f b, v8f c) {
  // (neg_a, A, neg_b, B, c_mod, C, reuse_a, reuse_b)
  return __builtin_amdgcn_wmma_f32_16x16x4_f32(false, a, false, b, (short)0, c,
                                               false, false);
}
#endif

// ---------------------------------------------------------------------------
// Weight transpose: in[K,N] -> out[N,K]  (tiny matrices, one-shot per call)
// Lets B fragments be loaded as a single contiguous global_load_b64 per tile.
// ---------------------------------------------------------------------------
__global__ void transpose_kernel(const float* __restrict__ in,
                                 float* __restrict__ out, int K, int N) {
  const int idx = blockIdx.x * 256 + threadIdx.x;
  if (idx >= K * N) return;
  const int k = idx / N, n = idx % N;
  out[(size_t)n * K + k] = in[idx];
}

// ---------------------------------------------------------------------------
// WMMA GEMM: C[M,N] = act(A[M,K] @ W[K,N] (+ bias)), W given transposed WT[N,K]
// One wave computes a 16x128 slab (8 accumulators), K stepped by 4.
// Inner loop: 1 A global_load_b64 + 8 B global_load_b64 -> 8 v_wmma.
// A frag: lane(0-15)=row, VGPR0/1 = K {0,1} (lanes 0-15) / {2,3} (lanes 16-31)
// B frag: VGPR0 = rows K{0,2}, VGPR1 = rows K{1,3}, cols striped across lanes
// Requires N % 128 == 0, K % 4 == 0.
// ---------------------------------------------------------------------------
template <int ACT>  // 0 = none, 1 = relu
__global__ __launch_bounds__(256) void gemm_kernel(
    const float* __restrict__ A, const float* __restrict__ WT,
    const float* __restrict__ bias, float* __restrict__ C, int M, int K, int N) {
  const int lane = threadIdx.x & 31;
  const int wave = threadIdx.x >> 5;
  const int row0 = (blockIdx.x * 8 + wave) * 16;
  if (row0 >= M) return;  // wave-uniform exit; no barriers in this kernel
  const int col0 = blockIdx.y * 128;
  const int r = lane & 15;
  const int hsel = lane >> 4;
  int arow = row0 + r;
  if (arow >= M) arow = M - 1;  // clamp loads; stores masked below

#ifdef HAVE_WMMA_F32
  const float* Ap = A + (size_t)arow * K + hsel * 2;
  const float* Bp = WT + (size_t)(col0 + r) * K + hsel * 2;
  const v8f zero = {};
  v8f acc[8] = {zero, zero, zero, zero, zero, zero, zero, zero};
  for (int k = 0; k < K; k += 4) {
    v2f a = *(const v2f*)(Ap + k);  // A[row][k+hsel*2 .. +1]
#pragma unroll
    for (int t = 0; t < 8; ++t) {
      v2f b = *(const v2f*)(Bp + (size_t)t * 16 * K + k);  // W[k..][col0+t*16+r]
      acc[t] = wmma4(a, b, acc[t]);
    }
  }
#pragma unroll
  for (int t = 0; t < 8; ++t) {
    const int col = col0 + t * 16 + r;
    const float bv = bias ? bias[col] : 0.0f;
#pragma unroll
    for (int v = 0; v < 8; ++v) {
      const int row = row0 + v + hsel * 8;
      if (row < M) {
        float val = acc[t][v] + bv;
        if (ACT == 1) val = fmaxf(val, 0.0f);
        C[(size_t)row * N + col] = val;
      }
    }
  }
#else
  // Naive fallback (host pass / missing builtin) — same output mapping.
  for (int t = 0; t < 8; ++t) {
    const int col = col0 + t * 16 + r;
    for (int v = 0; v < 8; ++v) {
      const int row = row0 + v + hsel * 8;
      if (row >= M) continue;
      float s = bias ? bias[col] : 0.0f;
      for (int k = 0; k < K; ++k)
        s += A[(size_t)row * K + k] * WT[(size_t)col * K + k];
      if (ACT == 1) s = fmaxf(s, 0.0f);
      C[(size_t)row * N + col] = s;
    }
  }
#endif
}

// ---------------------------------------------------------------------------
// Degree / normalization (layer-invariant, computed once)
// ---------------------------------------------------------------------------
__global__ void fill_kernel(float* __restrict__ p, float v, int n) {
  int i = blockIdx.x * blockDim.x + threadIdx.x;
  if (i < n) p[i] = v;
}

__global__ void deg_count_kernel(float* __restrict__ deg, const int* __restrict__ dst,
                                 int E) {
  int e = blockIdx.x * blockDim.x + threadIdx.x;
  if (e < E) unsafeAtomicAdd(&deg[dst[e]], 1.0f);
}

__global__ void rsqrt_kernel(const float* __restrict__ deg, float* __restrict__ dinv,
                             int n) {
  int i = blockIdx.x * blockDim.x + threadIdx.x;
  if (i < n) dinv[i] = rsqrtf(deg[i]);  // deg >= 1 always (self loops)
}

// ---------------------------------------------------------------------------
// GCN aggregation: agg[i] = xw[i]*dinv[i]^2 (self loop), then atomic scatter
// agg[dst] += xw[src] * dinv[src]*dinv[dst]
// ---------------------------------------------------------------------------
__global__ void agg_init_kernel(const float* __restrict__ xw,
                                const float* __restrict__ dinv,
                                float* __restrict__ agg, int n) {
  size_t idx = (size_t)blockIdx.x * blockDim.x + threadIdx.x;
  if (idx >= (size_t)n * HDIM) return;
  const int i = (int)(idx >> 8);  // HDIM == 256
  const float d = dinv[i];
  agg[idx] = xw[idx] * d * d;
}

// 4 edges per block; 64 lanes per edge; float4-wide loads, 4 atomics each.
__global__ __launch_bounds__(256) void agg_edge_kernel(
    const float* __restrict__ xw, const float* __restrict__ dinv,
    const int* __restrict__ src, const int* __restrict__ dst,
    float* __restrict__ agg, int E) {
  const int e = blockIdx.x * 4 + (threadIdx.x >> 6);
  if (e >= E) return;
  const int j4 = threadIdx.x & 63;  // float4 index within row (64*4 = 256)
  const int s = src[e], d = dst[e];
  const float norm = dinv[s] * dinv[d];
  const float4 v = ((const float4*)(xw + (size_t)s * HDIM))[j4];
  float* ap = agg + (size_t)d * HDIM + j4 * 4;
  unsafeAtomicAdd(ap + 0, v.x * norm);
  unsafeAtomicAdd(ap + 1, v.y * norm);
  unsafeAtomicAdd(ap + 2, v.z * norm);
  unsafeAtomicAdd(ap + 3, v.w * norm);
}

// ---------------------------------------------------------------------------
// out = relu(layernorm(agg + b)) + res    (one wave per node, H=256 -> 8/lane)
// ---------------------------------------------------------------------------
__global__ __launch_bounds__(256) void ln_relu_res_kernel(
    const float* __restrict__ agg, const float* __restrict__ bias,
    const float* __restrict__ gamma, const float* __restrict__ beta,
    const float* __restrict__ res, float* __restrict__ out, int n) {
  const int lane = threadIdx.x & 31;
  const int node = blockIdx.x * 8 + (threadIdx.x >> 5);
  if (node >= n) return;
  const float* ap = agg + (size_t)node * HDIM;
  float v[8];
  float s = 0.0f;
#pragma unroll
  for (int i = 0; i < 8; ++i) {
    const int j = lane + i * 32;
    v[i] = ap[j] + bias[j];
    s += v[i];
  }
#pragma unroll
  for (int m = 16; m >= 1; m >>= 1) s += __shfl_xor(s, m, 32);
  const float mu = s * (1.0f / HDIM);
  float q = 0.0f;
#pragma unroll
  for (int i = 0; i < 8; ++i) {
    const float d = v[i] - mu;
    q += d * d;
  }
#pragma unroll
  for (int m = 16; m >= 1; m >>= 1) q += __shfl_xor(q, m, 32);
  const float rs = rsqrtf(q * (1.0f / HDIM) + 1e-5f);
  const float* rp = res + (size_t)node * HDIM;
  float* op = out + (size_t)node * HDIM;
#pragma unroll
  for (int i = 0; i < 8; ++i) {
    const int j = lane + i * 32;
    const float o = (v[i] - mu) * rs * gamma[j] + beta[j];
    op[j] = fmaxf(o, 0.0f) + rp[j];
  }
}

// ---------------------------------------------------------------------------
// Fused edge predictor: per wave, 16 edges through
//   h1 = relu(concat(x[src],x[dst]) @ W1[512,256] + b1)   (WMMA, gathered A)
//   h2 = relu(h1 @ W2[256,128] + b2)                      (WMMA, A from LDS)
//   out = softplus(h2 @ W3[128,2] + b3)                   (per-lane dot)
// W1/W2 passed transposed. Block = 64 threads (2 waves), no __syncthreads.
// ---------------------------------------------------------------------------
__global__ __launch_bounds__(64) void edge_mlp_kernel(
    const float* __restrict__ x, const int* __restrict__ src,
    const int* __restrict__ dst, const float* __restrict__ W1T /*[256,512]*/,
    const float* __restrict__ b1, const float* __restrict__ W2T /*[128,256]*/,
    const float* __restrict__ b2, const float* __restrict__ W3 /*[128,2]*/,
    const float* __restrict__ b3, float* __restrict__ out, int E) {
  __shared__ float h1[2][16][260];  // +4 pad
  __shared__ float h2[2][16][132];
  const int lane = threadIdx.x & 31;
  const int wave = threadIdx.x >> 5;
  const int e0 = (blockIdx.x * 2 + wave) * 16;
  if (e0 >= E) return;  // wave-uniform
  const int r = lane & 15;
  const int hsel = lane >> 4;
  int e = e0 + r;
  if (e >= E) e = E - 1;
  const float* ps = x + (size_t)src[e] * HDIM;
  const float* pd = x + (size_t)dst[e] * HDIM;

#ifdef HAVE_WMMA_F32
  const v8f zero = {};
  // ---- layer 1: K=512 (concat), N=256 in 2 slabs of 128 ----
  for (int np = 0; np < 2; ++np) {
    const int col0 = np * 128;
    const float* bp = W1T + (size_t)(col0 + r) * 512 + hsel * 2;
    v8f acc[8] = {zero, zero, zero, zero, zero, zero, zero, zero};
    for (int k = 0; k < 2 * HDIM; k += 4) {
      const int kk = k + hsel * 2;
      const float* ap = (kk < HDIM) ? (ps + kk) : (pd + (kk - HDIM));
      v2f a = *(const v2f*)ap;
#pragma unroll
      for (int t = 0; t < 8; ++t) {
        v2f b = *(const v2f*)(bp + (size_t)t * 16 * 512 + k);
        acc[t] = wmma4(a, b, acc[t]);
      }
    }
#pragma unroll
    for (int t = 0; t < 8; ++t) {
      const int col = col0 + t * 16 + r;
      const float bv = b1[col];
#pragma unroll
      for (int v = 0; v < 8; ++v)
        h1[wave][v + hsel * 8][col] = fmaxf(acc[t][v] + bv, 0.0f);
    }
  }
  // ---- layer 2: K=256, N=128 in a single 128-wide slab ----
  {
    const float* bp = W2T + (size_t)r * 256 + hsel * 2;
    v8f acc[8] = {zero, zero, zero, zero, zero, zero, zero, zero};
    for (int k = 0; k < HDIM; k += 4) {
      v2f a = *(const v2f*)&h1[wave][r][k + hsel * 2];
#pragma unroll
      for (int t = 0; t < 8; ++t) {
        v2f b = *(const v2f*)(bp + (size_t)t * 16 * 256 + k);
        acc[t] = wmma4(a, b, acc[t]);
      }
    }
#pragma unroll
    for (int t = 0; t < 8; ++t) {
      const int col = t * 16 + r;
      const float bv = b2[col];
#pragma unroll
      for (int v = 0; v < 8; ++v)
        h2[wave][v + hsel * 8][col] = fmaxf(acc[t][v] + bv, 0.0f);
    }
  }
#else
  // Naive fallback (host pass / missing builtin)
  for (int mm = 0; mm < 16; ++mm) {
    int ec = e0 + mm;
    if (ec >= E) ec = E - 1;
    const float* qs = x + (size_t)src[ec] * HDIM;
    const float* qd = x + (size_t)dst[ec] * HDIM;
    for (int col = lane; col < HDIM; col += 32) {
      float s = b1[col];
      for (int k = 0; k < HDIM; ++k) s += qs[k] * W1T[(size_t)col * 512 + k];
      for (int k = 0; k < HDIM; ++k) s += qd[k] * W1T[(size_t)col * 512 + HDIM + k];
      h1[wave][mm][col] = fmaxf(s, 0.0f);
    }
  }
  for (int mm = 0; mm < 16; ++mm)
    for (int col = lane; col < 128; col += 32) {
      float s = b2[col];
      for (int k = 0; k < HDIM; ++k) s += h1[wave][mm][k] * W2T[(size_t)col * 256 + k];
      h2[wave][mm][col] = fmaxf(s, 0.0f);
    }
#endif
  // ---- layer 3: N=2; lanes 0-15 -> col 0, lanes 16-31 -> col 1 ----
  const int m = r, j = hsel;
  float z = b3[j];
  for (int k = 0; k < 128; ++k) z += h2[wave][m][k] * W3[k * 2 + j];
  const float p = (z > 0.0f) ? (z + log1pf(expf(-z))) : log1pf(expf(z));
  const int ee = e0 + m;
  if (ee < E) out[(size_t)ee * 2 + j] = p;
}

// ---------------------------------------------------------------------------
// Host orchestration
// ---------------------------------------------------------------------------
extern "C" void kernel_launch(void* const* d_in, const int* in_sizes, int n_in,
                              void* d_out, int out_size, void* d_ws, size_t ws_size,
                              hipStream_t stream) {
  (void)in_sizes; (void)n_in; (void)out_size; (void)ws_size;
  // JAX pytree flat order (dict keys sorted):
  // 0:node_features 1:edge_features 2:edge_index
  // 3:ee1_w 4:ee1_b 5:ee2_w 6:ee2_b                (edge encoder: dead code)
  // 7..14: gcn (W,b) x4    15..22: ln (gamma,beta) x4
  // 23:ne1_w 24:ne1_b 25:ne2_w 26:ne2_b
  // 27:pp1_w 28:pp1_b 29:pp2_w 30:pp2_b 31:pp3_w 32:pp3_b
  const float* nf = (const float*)d_in[0];
  const int* eidx = (const int*)d_in[2];
  const int* src = eidx;
  const int* dst = eidx + NEDGES;
  const float* gcnW[4], *gcnB[4], *lnG[4], *lnB[4];
  for (int l = 0; l < 4; ++l) {
    gcnW[l] = (const float*)d_in[7 + 2 * l];
    gcnB[l] = (const float*)d_in[8 + 2 * l];
    lnG[l]  = (const float*)d_in[15 + 2 * l];
    lnB[l]  = (const float*)d_in[16 + 2 * l];
  }
  const float* ne1w = (const float*)d_in[23];
  const float* ne1b = (const float*)d_in[24];
  const float* ne2w = (const float*)d_in[25];
  const float* ne2b = (const float*)d_in[26];
  const float* pp1w = (const float*)d_in[27];
  const float* pp1b = (const float*)d_in[28];
  const float* pp2w = (const float*)d_in[29];
  const float* pp2b = (const float*)d_in[30];
  const float* pp3w = (const float*)d_in[31];
  const float* pp3b = (const float*)d_in[32];

  const size_t NH = (size_t)NNODES * HDIM;
  float* ws   = (float*)d_ws;
  float* x    = ws;            // [N,H] current node state
  float* t    = ws + NH;       // [N,H] encoder tmp / per-layer xw
  float* agg  = ws + 2 * NH;   // [N,H] aggregation buffer
  float* deg  = ws + 3 * NH;   // [N]
  float* dinv = deg + NNODES;  // [N]
  // transposed weights
  float* wt = dinv + NNODES;
  float* ne1T = wt;                 wt += (size_t)HDIM * DNODE;   // [256,64]
  float* ne2T = wt;                 wt += (size_t)HDIM * HDIM;    // [256,256]
  float* gcnT[4];
  for (int l = 0; l < 4; ++l) { gcnT[l] = wt; wt += (size_t)HDIM * HDIM; }
  float* pp1T = wt;                 wt += (size_t)HDIM * 2 * HDIM; // [256,512]
  float* pp2T = wt;                 wt += (size_t)128 * HDIM;      // [128,256]

  float* xout = (float*)d_out;  // final x  [N,H]
  float* par  = xout + NH;      // parasitics [E,2]

  const dim3 blk(256);
  const int rowTiles = (NNODES + 15) / 16;
  const dim3 gGemm((rowTiles + 7) / 8, HDIM / 128);

  // one-shot weight transposes (tiny)
  transpose_kernel<<<(DNODE * HDIM + 255) / 256, blk, 0, stream>>>(ne1w, ne1T, DNODE, HDIM);
  transpose_kernel<<<(HDIM * HDIM + 255) / 256, blk, 0, stream>>>(ne2w, ne2T, HDIM, HDIM);
  for (int l = 0; l < 4; ++l)
    transpose_kernel<<<(HDIM * HDIM + 255) / 256, blk, 0, stream>>>(gcnW[l], gcnT[l], HDIM, HDIM);
  transpose_kernel<<<(2 * HDIM * HDIM + 255) / 256, blk, 0, stream>>>(pp1w, pp1T, 2 * HDIM, HDIM);
  transpose_kernel<<<(HDIM * 128 + 255) / 256, blk, 0, stream>>>(pp2w, pp2T, HDIM, 128);

  // node encoder: x = relu(nf@ne1 + b) @ ne2 + b
  gemm_kernel<1><<<gGemm, blk, 0, stream>>>(nf, ne1T, ne1b, t, NNODES, DNODE, HDIM);
  gemm_kernel<0><<<gGemm, blk, 0, stream>>>(t, ne2T, ne2b, x, NNODES, HDIM, HDIM);

  // symmetric normalization (self loops -> deg starts at 1)
  fill_kernel<<<(NNODES + 255) / 256, blk, 0, stream>>>(deg, 1.0f, NNODES);
  deg_count_kernel<<<(NEDGES + 255) / 256, blk, 0, stream>>>(deg, dst, NEDGES);
  rsqrt_kernel<<<(NNODES + 255) / 256, blk, 0, stream>>>(deg, dinv, NNODES);

  // 4 GCN layers
  for (int l = 0; l < 4; ++l) {
    gemm_kernel<0><<<gGemm, blk, 0, stream>>>(x, gcnT[l], nullptr, t, NNODES, HDIM, HDIM);
    agg_init_kernel<<<(int)((NH + 255) / 256), blk, 0, stream>>>(t, dinv, agg, NNODES);
    agg_edge_kernel<<<(NEDGES + 3) / 4, blk, 0, stream>>>(t, dinv, src, dst, agg, NEDGES);
    float* o = (l == 3) ? xout : x;  // final layer writes straight to d_out
    ln_relu_res_kernel<<<(NNODES + 7) / 8, blk, 0, stream>>>(agg, gcnB[l], lnG[l],
                                                             lnB[l], x, o, NNODES);
  }

  // fused edge MLP -> parasitics
  edge_mlp_kernel<<<(NEDGES + 31) / 32, dim3(64), 0, stream>>>(
      xout, src, dst, pp1T, pp1b, pp2T, pp2b, pp3w, pp3b, par, NEDGES);
}